// LaplacianSmoothingLoss_197568495958
// MI455X (gfx1250) — compile-verified
//
#include <hip/hip_runtime.h>
#include <hip/hip_bf16.h>

// Vector types for the CDNA5 WMMA f32 16x16x4 builtin:
// A = 16x4 f32  -> 64 elems / 32 lanes = 2 VGPRs  -> v2f
// B = 4x16 f32  -> 64 elems / 32 lanes = 2 VGPRs  -> v2f
// C/D = 16x16 f32 -> 256 elems / 32 lanes = 8 VGPRs -> v8f
typedef float v2f __attribute__((ext_vector_type(2)));
typedef float v8f __attribute__((ext_vector_type(8)));

#define THREADS 256

// ---------------------------------------------------------------------------
// Phase 0: zero the scatter accumulators (s[NV*3] ++ cnt[NV]) each launch so
// the kernel is deterministic across graph replays.
// ---------------------------------------------------------------------------
__global__ void lsl_zero_kernel(float* __restrict__ p, long long n) {
  long long i = (long long)blockIdx.x * blockDim.x + threadIdx.x;
  long long stride = (long long)gridDim.x * blockDim.x;
  for (; i < n; i += stride) p[i] = 0.0f;
}

// ---------------------------------------------------------------------------
// Phase 1: edge scatter.  s[row] += 100*x[col]; cnt[row] += 1.
// Index/gather traffic is L2-resident on MI455X (192 MB L2 >> 56 MB hot set),
// so this phase is atomic-throughput bound.  Prefetch the sequential index
// streams (emits global_prefetch_b8 on gfx1250).
// ---------------------------------------------------------------------------
__global__ void lsl_scatter_kernel(const float* __restrict__ x,
                                   const int* __restrict__ row,
                                   const int* __restrict__ col,
                                   float* __restrict__ s,
                                   float* __restrict__ cnt,
                                   int ne) {
  int e = blockIdx.x * blockDim.x + threadIdx.x;
  if (e >= ne) return;
  if (e + 8192 < ne) {
    __builtin_prefetch(&row[e + 8192], 0, 1);
    __builtin_prefetch(&col[e + 8192], 0, 1);
  }
  int r = row[e];
  int c = col[e];
  float x0 = x[3 * c + 0] * 100.0f;
  float x1 = x[3 * c + 1] * 100.0f;
  float x2 = x[3 * c + 2] * 100.0f;
  atomicAdd(&s[3 * r + 0], x0);
  atomicAdd(&s[3 * r + 1], x1);
  atomicAdd(&s[3 * r + 2], x2);
  atomicAdd(&cnt[r], 1.0f);
}

// ---------------------------------------------------------------------------
// Phase 2: per-vertex loss + WMMA block reduction.
//
// Each thread computes its per-vertex partial; 256 partials land in LDS.
// Wave 0 (EXEC all-ones inside the wave-uniform branch) reduces them with
// V_WMMA_F32_16X16X4_F32:  D = A x B + C with B == all-ones gives
// D[m][n] = rowsum(A[m]) + C[m][n], so 4 accumulating WMMAs fold all 256
// partials into C.  Column N=0 of C sits in lane 0 (M=0..7, VGPR 0..7) and
// lane 16 (M=8..15, VGPR 0..7); one shuffle finishes the sum.
// ---------------------------------------------------------------------------
__global__ void lsl_finalize_kernel(const float* __restrict__ x,
                                    const float* __restrict__ xyz_dif,
                                    const float* __restrict__ s,
                                    const float* __restrict__ cnt,
                                    float* __restrict__ block_sums,
                                    int nv) {
  __shared__ float partial[THREADS];
  const int tid = threadIdx.x;
  const int v = blockIdx.x * blockDim.x + tid;

  float local = 0.0f;
  if (v < nv) {
    float c = fmaxf(cnt[v], 1.0f);
    float m0 = s[3 * v + 0] / c;
    float m1 = s[3 * v + 1] / c;
    float m2 = s[3 * v + 2] / c;
    float nrm = sqrtf(m0 * m0 + m1 * m1 + m2 * m2);
    float k = 100.0f / nrm;   // matches reference: /norm * 100 (0 -> NaN)
    m0 *= k; m1 *= k; m2 *= k;
    float d0 = fabsf(fabsf(x[3 * v + 0] * 100.0f - m0) - xyz_dif[3 * v + 0]);
    float d1 = fabsf(fabsf(x[3 * v + 1] * 100.0f - m1) - xyz_dif[3 * v + 1]);
    float d2 = fabsf(fabsf(x[3 * v + 2] * 100.0f - m2) - xyz_dif[3 * v + 2]);
    local = d0 + d1 + d2;
  }
  partial[tid] = local;
  __syncthreads();

  if (tid < 32) {            // wave 0 only: EXEC all 1s within the wave
    v8f acc = {};
    v2f ones;
    ones.x = 1.0f;
    ones.y = 1.0f;
#pragma unroll
    for (int ch = 0; ch < 4; ++ch) {
      v2f a;
      a.x = partial[ch * 64 + tid];
      a.y = partial[ch * 64 + 32 + tid];
      // (neg_a, A, neg_b, B, c_mod, C, reuse_a, reuse_b)
      acc = __builtin_amdgcn_wmma_f32_16x16x4_f32(
          false, a, false, ones, (short)0, acc, false, false);
    }
    float t = acc[0] + acc[1] + acc[2] + acc[3] +
              acc[4] + acc[5] + acc[6] + acc[7];
    float t_hi = __shfl(t, 16, 32);   // lane 0 pulls lane 16 (M=8..15 half)
    if (tid == 0) block_sums[blockIdx.x] = t + t_hi;
  }
}

// ---------------------------------------------------------------------------
// Phase 3: deterministic final mean (fixed-order double accumulation).
// ---------------------------------------------------------------------------
__global__ void lsl_mean_kernel(const float* __restrict__ block_sums,
                                int nblocks, float* __restrict__ out,
                                long long count) {
  if (blockIdx.x == 0 && threadIdx.x == 0) {
    double acc = 0.0;
    for (int i = 0; i < nblocks; ++i) acc += (double)block_sums[i];
    out[0] = (float)(acc / (double)count);
  }
}

extern "C" void kernel_launch(void* const* d_in, const int* in_sizes, int n_in,
                              void* d_out, int out_size, void* d_ws, size_t ws_size,
                              hipStream_t stream) {
  const float* x       = (const float*)d_in[0];   // [NV,3]
  const float* xyz_dif = (const float*)d_in[1];   // [NV,3]
  const int*   row     = (const int*)d_in[2];     // [NE]
  const int*   col     = (const int*)d_in[3];     // [NE]
  float*       out     = (float*)d_out;           // scalar

  const int nv = in_sizes[0] / 3;
  const int ne = in_sizes[2];

  // Workspace layout: s[nv*3] | cnt[nv] | block_sums[nblocksF]
  float* s   = (float*)d_ws;
  float* cnt = s + (size_t)nv * 3;
  const int nblocksF = (nv + THREADS - 1) / THREADS;
  float* block_sums = cnt + nv;

  const long long nzero = (long long)nv * 4;      // s + cnt
  lsl_zero_kernel<<<2048, THREADS, 0, stream>>>(s, nzero);

  const int nblocksE = (ne + THREADS - 1) / THREADS;
  lsl_scatter_kernel<<<nblocksE, THREADS, 0, stream>>>(x, row, col, s, cnt, ne);

  lsl_finalize_kernel<<<nblocksF, THREADS, 0, stream>>>(x, xyz_dif, s, cnt,
                                                        block_sums, nv);

  lsl_mean_kernel<<<1, 32, 0, stream>>>(block_sums, nblocksF, out,
                                        (long long)nv * 3);
}